// Quantizer_27754078667208
// MI455X (gfx1250) — compile-verified
//
#include <hip/hip_runtime.h>
#include <stdint.h>

// Problem geometry (fixed by the reference).
#define BDIM 4
#define SDIM 2048
#define HDIM 4096
#define NROWS (BDIM * SDIM)          // 8192 tokens
#define THREADS 128                  // 4 wave32s per workgroup
#define CHUNKS 8                     // float4 chunks per thread per row: 4096/(128*4)
#define ROWS_PER_WG 8
#define NUM_WG (NROWS / ROWS_PER_WG) // 1024

#define CLIP_OVER_QMAX (0.9f / 7.0f)

// Issue one 16B async DMA: global -> LDS (CDNA5 GLOBAL_LOAD_ASYNC_TO_LDS_B128, ASYNCcnt-tracked).
__device__ __forceinline__ void async_g2lds_b128(uint32_t lds_byte_addr, const void* gaddr) {
    asm volatile("global_load_async_to_lds_b128 %0, %1, off"
                 :: "v"(lds_byte_addr), "v"(gaddr)
                 : "memory");
}

__global__ __launch_bounds__(THREADS)
void quant_int4_pack_kernel(const float* __restrict__ x,
                            uint16_t* __restrict__ packed,   // 2 bytes = 4 source elems
                            float* __restrict__ scales) {
    __shared__ float buf[2][HDIM];   // 2 x 16 KiB double buffer (DMA landing zone)
    __shared__ float wmax[4];        // per-wave partial absmax

    const int t    = threadIdx.x;
    const int wave = t >> 5;
    const int lane = t & 31;
    const int row0 = blockIdx.x * ROWS_PER_WG;

    // Issue the 8 per-thread b128 DMAs that stage one full row into buf[b].
    auto issue_row = [&](int row, int b) {
        const float4* src = (const float4*)(x + (size_t)row * HDIM);
        const uint32_t lds0 = (uint32_t)(uintptr_t)(&buf[b][0]);
        #pragma unroll
        for (int j = 0; j < CHUNKS; ++j) {
            const int idx = j * THREADS + t;                 // float4 index within row
            async_g2lds_b128(lds0 + (uint32_t)idx * 16u, (const void*)(src + idx));
        }
    };

    // Prime the pipeline.
    issue_row(row0, 0);

    for (int i = 0; i < ROWS_PER_WG; ++i) {
        const int row = row0 + i;
        const int b   = i & 1;

        // Overlap: kick off next row's DMA before consuming this one.
        if (i + 1 < ROWS_PER_WG) {
            issue_row(row + 1, b ^ 1);
            // 8 async issues/wave/row, in-order completion: <=8 outstanding
            // means the current row's 8 transfers have landed in LDS.
            asm volatile("s_wait_asynccnt 8" ::: "memory");
        } else {
            asm volatile("s_wait_asynccnt 0" ::: "memory");
        }
        __syncthreads();   // buf[b] visible to all 4 waves

        // Pass 1: LDS -> registers once; local absmax over this thread's 32 elems.
        const float4* lbuf = (const float4*)&buf[b][0];
        float4 v[CHUNKS];
        float m = 0.0f;
        #pragma unroll
        for (int j = 0; j < CHUNKS; ++j) {
            v[j] = lbuf[j * THREADS + t];
            m = fmaxf(m, fmaxf(fmaxf(fabsf(v[j].x), fabsf(v[j].y)),
                               fmaxf(fabsf(v[j].z), fabsf(v[j].w))));
        }

        // wave32 reduction, then cross-wave via LDS.
        #pragma unroll
        for (int off = 16; off >= 1; off >>= 1)
            m = fmaxf(m, __shfl_xor(m, off, 32));
        if (lane == 0) wmax[wave] = m;
        __syncthreads();
        const float am = fmaxf(fmaxf(wmax[0], wmax[1]), fmaxf(wmax[2], wmax[3]));

        const float scale = fmaxf(am * CLIP_OVER_QMAX, 1e-8f);
        const float inv   = 1.0f / scale;

        // Pass 2: quantize from registers, pack nibbles, coalesced u16 stores.
        uint16_t* orow = packed + (size_t)row * (HDIM / 4);  // HDIM/2 bytes = HDIM/4 u16
        #pragma unroll
        for (int j = 0; j < CHUNKS; ++j) {
            // rintf == round-half-to-even (matches jnp.round), clamp to int4 range.
            const int q0 = (int)fminf(fmaxf(rintf(v[j].x * inv), -8.0f), 7.0f);
            const int q1 = (int)fminf(fmaxf(rintf(v[j].y * inv), -8.0f), 7.0f);
            const int q2 = (int)fminf(fmaxf(rintf(v[j].z * inv), -8.0f), 7.0f);
            const int q3 = (int)fminf(fmaxf(rintf(v[j].w * inv), -8.0f), 7.0f);
            const uint32_t b0 = (uint32_t)(q0 & 0xF) | ((uint32_t)(q1 & 0xF) << 4);
            const uint32_t b1 = (uint32_t)(q2 & 0xF) | ((uint32_t)(q3 & 0xF) << 4);
            orow[j * THREADS + t] = (uint16_t)(b0 | (b1 << 8));  // little-endian: elem0 low nibble
        }
        if (t == 0) scales[row] = scale;
        // loop-end: next iteration's barrier (after s_wait_asynccnt) fences buffer reuse.
    }
}

extern "C" void kernel_launch(void* const* d_in, const int* in_sizes, int n_in,
                              void* d_out, int out_size, void* d_ws, size_t ws_size,
                              hipStream_t stream) {
    const float* x = (const float*)d_in[0];
    // Tuple outputs concatenated flat in return order:
    //   [0, 16777216) bytes : packed int8 (4,2048,2048)
    //   then 8192 f32 scales (4,2048,1)
    uint16_t* packed = (uint16_t*)d_out;
    float* scales = (float*)((char*)d_out + (size_t)NROWS * (HDIM / 2));

    quant_int4_pack_kernel<<<dim3(NUM_WG), dim3(THREADS), 0, stream>>>(x, packed, scales);
}